// YOLOv2Loss_22462678958722
// MI455X (gfx1250) — compile-verified
//
#include <hip/hip_runtime.h>
#include <math.h>

#define AA 5
#define CC 80
#define TT 50
#define HF 38
#define WF 38
#define BB 64
#define CELLS (HF * WF)                 // 1444
#define META_STRIDE 16
#define META_PER_B (TT * META_STRIDE)   // 800 floats per batch
#define META_FLOATS (BB * META_PER_B)   // 51200 floats
#define NBLK (BB * AA)                  // 320 main blocks
#define PART_STRIDE 512

__constant__ float c_AW[5] = {0.57273f, 1.87446f, 3.33843f, 7.88282f, 9.77052f};
__constant__ float c_AH[5] = {0.677385f, 2.06253f, 5.47434f, 3.52778f, 9.16828f};

__device__ inline float iou_fn(float cx1, float cy1, float w1, float h1,
                               float cx2, float cy2, float w2, float h2) {
  float mx = fminf(cx1 - w1 * 0.5f, cx2 - w2 * 0.5f);
  float Mx = fmaxf(cx1 + w1 * 0.5f, cx2 + w2 * 0.5f);
  float my = fminf(cy1 - h1 * 0.5f, cy2 - h2 * 0.5f);
  float My = fmaxf(cy1 + h1 * 0.5f, cy2 + h2 * 0.5f);
  float cw = w1 + w2 - (Mx - mx);
  float ch = h1 + h2 - (My - my);
  float inter = (cw > 0.f && ch > 0.f) ? cw * ch : 0.f;
  float uni = w1 * h1 + w2 * h2 - inter;
  return (uni > 0.f) ? inter / uni : 0.f;
}

__device__ inline float sigm(float x) { return 1.f / (1.f + expf(-x)); }

// ---- CDNA5 async global->LDS copy (ISA §10.7 / §15.18 opcode 96), ASYNCcnt tracked
__device__ inline void async_g2l_b32(const float* gsrc, const float* ldst) {
  unsigned lds_off = (unsigned)(unsigned long long)(uintptr_t)ldst;
  unsigned long long ga = (unsigned long long)(uintptr_t)gsrc;
  asm volatile("global_load_async_to_lds_b32 %0, %1, off"
               :: "v"(lds_off), "v"(ga) : "memory");
}
__device__ inline void wait_asynccnt0() {
#if __has_builtin(__builtin_amdgcn_s_wait_asynccnt)
  __builtin_amdgcn_s_wait_asynccnt(0);
#else
  asm volatile("s_wait_asynccnt 0" ::: "memory");
#endif
}

// ============ Kernel 1: per-GT metadata (B blocks x 64 threads) ============
// meta[b][t][0..15] = gx,gy,gw,gh, tx,ty,tw,th, cls, valid, gi, gj, best_n, 0,0,0
__global__ void __launch_bounds__(64)
k_targets(const float* __restrict__ tgt, float* __restrict__ meta) {
  __shared__ float scx[TT];
  const int b = blockIdx.x;
  const int t = threadIdx.x;
  float cls = 0.f, cx = 0.f, cy = 0.f, wn = 0.f, hn = 0.f;
  if (t < TT) {
    const float* p = tgt + (size_t)b * (TT * 5) + t * 5;
    cls = p[0]; cx = p[1]; cy = p[2]; wn = p[3]; hn = p[4];
    scx[t] = cx;
  }
  __syncthreads();
  if (t < TT) {
    // cumprod validity: valid iff no cx == -1 at or before t
    bool valid = true;
    for (int s = 0; s <= t; ++s) valid = valid && (scx[s] != -1.0f);
    float gx = cx * (float)WF, gy = cy * (float)HF;
    float gw = wn * (float)WF, gh = hn * (float)HF;
    // best anchor by center-0 IoU (first max wins, like jnp.argmax)
    int bn = 0; float best = -1.f;
    for (int n = 0; n < AA; ++n) {
      float iou = iou_fn(0.f, 0.f, gw, gh, 0.f, 0.f, c_AW[n], c_AH[n]);
      if (iou > best) { best = iou; bn = n; }
    }
    int gi = (int)gx; gi = gi < 0 ? 0 : (gi > WF - 1 ? WF - 1 : gi);
    int gj = (int)gy; gj = gj < 0 ? 0 : (gj > HF - 1 ? HF - 1 : gj);
    float txv = gx - (float)gi;
    float tyv = gy - (float)gj;
    float twv = valid ? logf(gw / c_AW[bn]) : 0.f;
    float thv = valid ? logf(gh / c_AH[bn]) : 0.f;
    float* m = meta + ((size_t)b * TT + t) * META_STRIDE;
    m[0] = gx;  m[1] = gy;  m[2] = gw;  m[3] = gh;
    m[4] = txv; m[5] = tyv; m[6] = twv; m[7] = thv;
    m[8] = cls; m[9] = valid ? 1.f : 0.f;
    m[10] = (float)gi; m[11] = (float)gj; m[12] = (float)bn;
    m[13] = 0.f; m[14] = 0.f; m[15] = 0.f;
  }
}

// ============ Kernel 2: per-cell loss (NBLK blocks x 256 threads) ============
__global__ void __launch_bounds__(256)
k_main(const float* __restrict__ pred, const float* __restrict__ meta,
       float* __restrict__ partials) {
  __shared__ float sm[META_PER_B];
  __shared__ float red[3][256];
  const int blk = blockIdx.x;
  const int b = blk / AA, a = blk % AA;
  const int tid = threadIdx.x;

  // stage this batch's GT table into LDS via async copies
  const float* g = meta + (size_t)b * META_PER_B;
  for (int i = tid; i < META_PER_B; i += 256) async_g2l_b32(g + i, &sm[i]);
  wait_asynccnt0();
  __syncthreads();

  const float aw = c_AW[a], ah = c_AH[a];
  const long long baseBA = ((long long)b * (AA * (5 + CC)) + (long long)a * (5 + CC)) * CELLS;

  float s_main = 0.f, s_cls = 0.f, s_cnt = 0.f;
  for (int cell = tid; cell < CELLS; cell += 256) {
    const int j = cell / WF, i = cell - j * WF;
    const float* pc = pred + baseBA + cell;
    const float xr = pc[0 * CELLS];
    const float yr = pc[1 * CELLS];
    const float wr = pc[2 * CELLS];
    const float hr = pc[3 * CELLS];
    const float cr = pc[4 * CELLS];
    const float sx = sigm(xr), sy = sigm(yr);
    const float bx = sx + (float)i, by = sy + (float)j;
    const float bw = expf(wr) * aw, bh = expf(hr) * ah;

    float maxiou = 0.f;
    int twin = -1;  // last GT targeting this cell wins (scatter last-wins)
    for (int t = 0; t < TT; ++t) {
      const float* m = &sm[t * META_STRIDE];
      if (m[9] != 0.f) {
        maxiou = fmaxf(maxiou, iou_fn(bx, by, bw, bh, m[0], m[1], m[2], m[3]));
        if ((int)m[12] == a && (int)m[10] == i && (int)m[11] == j) twin = t;
      }
    }
    const float conf = sigm(cr);
    if (twin >= 0) {
      const float* m = &sm[twin * META_STRIDE];
      const float tconf = iou_fn(m[0], m[1], m[2], m[3], bx, by, bw, bh);
      const float dx = sx - m[4], dy = sy - m[5];
      const float dw = wr - m[6], dh = hr - m[7];
      s_main += dx * dx + dy * dy + dw * dw + dh * dh;          // coord (scale 1)
      s_main += 5.f * (conf - tconf) * (conf - tconf);          // obj conf (scale 5)
      // class cross-entropy via log-softmax over 80 logits
      const int cstar = (int)m[8];
      float mx = -3.4e38f;
      for (int k = 0; k < CC; ++k) mx = fmaxf(mx, pc[(5 + k) * CELLS]);
      float se = 0.f;
      for (int k = 0; k < CC; ++k) se += expf(pc[(5 + k) * CELLS] - mx);
      const float lse = logf(se) + mx;
      s_cls += -(pc[(5 + cstar) * CELLS] - lse);
      s_cnt += 1.f;
    } else if (maxiou <= 0.6f) {
      s_main += conf * conf;  // noobj: tconf==0, mask==1
    }
  }

  red[0][tid] = s_main; red[1][tid] = s_cls; red[2][tid] = s_cnt;
  __syncthreads();
  for (int off = 128; off > 0; off >>= 1) {
    if (tid < off) {
      red[0][tid] += red[0][tid + off];
      red[1][tid] += red[1][tid + off];
      red[2][tid] += red[2][tid + off];
    }
    __syncthreads();
  }
  if (tid == 0) {
    partials[0 * PART_STRIDE + blk] = red[0][0];
    partials[1 * PART_STRIDE + blk] = red[1][0];
    partials[2 * PART_STRIDE + blk] = red[2][0];
  }
}

// ============ Kernel 3: WMMA-based final reduction (1 block x 32 threads) ====
typedef float v2f __attribute__((ext_vector_type(2)));
typedef float v8f __attribute__((ext_vector_type(8)));

__global__ void __launch_bounds__(32)
k_final(const float* __restrict__ partials, float* __restrict__ out) {
  const int lane = threadIdx.x;  // full wave32, EXEC all ones (WMMA requirement)
  float tot[3];
  for (int comp = 0; comp < 3; ++comp) {
    const float* p = partials + comp * PART_STRIDE;
#if __has_builtin(__builtin_amdgcn_wmma_f32_16x16x4_f32)
    // D(16x16) = A(16x4 values) x B(4x16 ones) + C: rows of D accumulate row
    // sums of A, so sum(all D entries) == 16 * sum(all A entries). 320 = 5*64.
    v8f c = {0.f, 0.f, 0.f, 0.f, 0.f, 0.f, 0.f, 0.f};
    v2f ones; ones.x = 1.f; ones.y = 1.f;
    for (int chunk = 0; chunk < 5; ++chunk) {
      const int idx = chunk * 64 + lane * 2;
      v2f av; av.x = p[idx]; av.y = p[idx + 1];
      c = __builtin_amdgcn_wmma_f32_16x16x4_f32(
          /*neg_a=*/false, av, /*neg_b=*/false, ones,
          /*c_mod=*/(short)0, c, /*reuse_a=*/false, /*reuse_b=*/false);
    }
    float s = c[0] + c[1] + c[2] + c[3] + c[4] + c[5] + c[6] + c[7];
    for (int off = 16; off > 0; off >>= 1) s += __shfl_xor(s, off, 32);
    tot[comp] = s * (1.0f / 16.0f);
#else
    float s = 0.f;
    for (int i = lane; i < NBLK; i += 32) s += p[i];
    for (int off = 16; off > 0; off >>= 1) s += __shfl_xor(s, off, 32);
    tot[comp] = s;
#endif
  }
  if (lane == 0) {
    const float nobj = fmaxf(tot[2], 1.0f);
    out[0] = tot[0] + tot[1] / nobj;
  }
}

extern "C" void kernel_launch(void* const* d_in, const int* in_sizes, int n_in,
                              void* d_out, int out_size, void* d_ws, size_t ws_size,
                              hipStream_t stream) {
  const float* pred = (const float*)d_in[0];   // (64, 425, 38, 38) f32
  const float* tgt  = (const float*)d_in[1];   // (64, 250) f32
  float* wsf = (float*)d_ws;
  float* meta = wsf;                        // 51200 floats
  float* partials = wsf + META_FLOATS;      // 3 * 512 floats

  k_targets<<<BB, 64, 0, stream>>>(tgt, meta);
  k_main<<<NBLK, 256, 0, stream>>>(pred, meta, partials);
  k_final<<<1, 32, 0, stream>>>(partials, (float*)d_out);
}